// GCN_2L_15857019257144
// MI455X (gfx1250) — compile-verified
//
#include <hip/hip_runtime.h>

// ---------------------------------------------------------------------------
// GCN 2-layer for MI455X (gfx1250): fp32 WMMA (16x16x4) GEMMs + L2-resident
// edge scatter with fp32 global atomics. Symmetric norm folded into GEMM
// epilogue (source side) and bias/ReLU pass (dest side).
// ---------------------------------------------------------------------------

typedef __attribute__((ext_vector_type(2))) float v2f;
typedef __attribute__((ext_vector_type(8))) float v8f;

#define N_NODES 50000
#define D_IN    128
#define H_DIM   128
#define D_OUT   64
#define N_EDGES 800000

// ---------------------------------------------------------------------------
// Degree / norm kernels
// ---------------------------------------------------------------------------
__global__ void k_init_deg(float* __restrict__ deg) {
  int i = blockIdx.x * blockDim.x + threadIdx.x;
  if (i < N_NODES) deg[i] = 1.0f;  // self loop contributes 1
}

__global__ void k_accum_deg(const long long* __restrict__ dst,
                            float* __restrict__ deg) {
  int e = blockIdx.x * blockDim.x + threadIdx.x;
  if (e < N_EDGES) atomicAdd(&deg[(int)dst[e]], 1.0f);
}

__global__ void k_deg_to_dinv(float* __restrict__ deg) {
  int i = blockIdx.x * blockDim.x + threadIdx.x;
  if (i < N_NODES) deg[i] = rsqrtf(deg[i]);  // deg >= 1 always
}

// ---------------------------------------------------------------------------
// WMMA GEMM: hs[i,:] = dinv[i] * (A[i,:] @ B)   (A: M x K, B: K x Nout)
// Also writes acc = hs (self-loop initialization of the aggregation buffer).
// One wave per 16x16 output tile. M, Nout, K all multiples of tile sizes.
// fp32 WMMA keeps full precision vs. the fp32 reference.
// ---------------------------------------------------------------------------
template <int K, int NOUT>
__global__ __launch_bounds__(128)
void k_gemm_scale(const float* __restrict__ A, const float* __restrict__ B,
                  const float* __restrict__ dinv, float* __restrict__ hs,
                  float* __restrict__ acc) {
  const int wave = threadIdx.x >> 5;
  const int lane = threadIdx.x & 31;
  const int tilesN = NOUT / 16;
  const int tile = blockIdx.x * 4 + wave;
  const int tm = tile / tilesN;
  const int tn = tile % tilesN;
  const int row0 = tm * 16;
  const int col0 = tn * 16;
  const int half = lane >> 4;   // 0: lanes 0-15, 1: lanes 16-31
  const int l    = lane & 15;

  // A layout (16x4 fp32, ISA 7.12.2): lane half 0 holds K=k..k+1 of row M=l,
  // lane half 1 holds K=k+2..k+3 of row M=l.
  const float* __restrict__ arow = A + (size_t)(row0 + l) * K;

  v8f c = {};
#pragma unroll 8
  for (int k = 0; k < K; k += 4) {
    const int kk = k + 2 * half;
    v2f a, b;
    a.x = arow[kk];
    a.y = arow[kk + 1];
    // B layout (4x16 fp32): v0 = row kk (N striped across lanes), v1 = row kk+1
    b.x = B[(size_t)kk * NOUT + col0 + l];
    b.y = B[(size_t)(kk + 1) * NOUT + col0 + l];
    // D = A*B + C ; emits v_wmma_f32_16x16x4_f32
    c = __builtin_amdgcn_wmma_f32_16x16x4_f32(
        /*neg_a=*/false, a, /*neg_b=*/false, b,
        /*c_mod=*/(short)0, c, /*reuse_a=*/false, /*reuse_b=*/false);
  }

  // C/D layout: VGPR v -> M = v (lanes 0-15) or v+8 (lanes 16-31), N = l.
#pragma unroll
  for (int v = 0; v < 8; ++v) {
    const int row = row0 + v + 8 * half;
    const float val = c[v] * dinv[row];
    const size_t idx = (size_t)row * NOUT + col0 + l;
    hs[idx]  = val;  // source-scaled features for edge gather
    acc[idx] = val;  // self-loop term initializes the accumulator
  }
}

// ---------------------------------------------------------------------------
// Edge scatter: acc[dst,:] += hs[src,:]. One thread per 4-float chunk;
// consecutive threads cover one edge's feature row -> coalesced gather and
// clustered atomics (L2-resident working set on the 192MB L2).
// ---------------------------------------------------------------------------
template <int DIM>
__global__ void k_scatter_edges(const long long* __restrict__ src,
                                const long long* __restrict__ dst,
                                const float* __restrict__ hs,
                                float* __restrict__ acc) {
  const int chunks = DIM / 4;
  const int t = blockIdx.x * blockDim.x + threadIdx.x;
  if (t >= N_EDGES * chunks) return;
  const int e  = t / chunks;       // power-of-two -> shift
  const int cc = t - e * chunks;
  const int s = (int)src[e];
  const int d = (int)dst[e];
  const float4 v = *(const float4*)(hs + (size_t)s * DIM + cc * 4);
  float* p = acc + (size_t)d * DIM + cc * 4;
  atomicAdd(p + 0, v.x);
  atomicAdd(p + 1, v.y);
  atomicAdd(p + 2, v.z);
  atomicAdd(p + 3, v.w);
}

// ---------------------------------------------------------------------------
// out[i,j] = relu(dinv[i] * acc[i,j] + bias[j])
// ---------------------------------------------------------------------------
template <int DIM>
__global__ void k_bias_relu(const float* __restrict__ acc,
                            const float* __restrict__ dinv,
                            const float* __restrict__ bias,
                            float* __restrict__ out) {
  const int t = blockIdx.x * blockDim.x + threadIdx.x;
  if (t >= N_NODES * DIM) return;
  const int i = t / DIM;           // power-of-two -> shift
  const int j = t & (DIM - 1);
  const float v = dinv[i] * acc[t] + bias[j];
  out[t] = v > 0.0f ? v : 0.0f;
}

// ---------------------------------------------------------------------------
// Launch
// ---------------------------------------------------------------------------
static inline size_t align256(size_t x) { return (x + 255) & ~(size_t)255; }

extern "C" void kernel_launch(void* const* d_in, const int* in_sizes, int n_in,
                              void* d_out, int out_size, void* d_ws, size_t ws_size,
                              hipStream_t stream) {
  const float*     x    = (const float*)d_in[0];
  const long long* ei   = (const long long*)d_in[1];  // [2, E] int64
  const float*     W1   = (const float*)d_in[2];      // [128,128]
  const float*     b1   = (const float*)d_in[3];      // [128]
  const float*     W2   = (const float*)d_in[4];      // [128,64]
  const float*     b2   = (const float*)d_in[5];      // [64]
  float*           out  = (float*)d_out;              // [N,64]

  const long long* src = ei;
  const long long* dst = ei + N_EDGES;

  // Workspace layout (~77 MB)
  char* ws = (char*)d_ws;
  float* dinv = (float*)ws;  ws += align256((size_t)N_NODES * 4);
  float* hs1  = (float*)ws;  ws += align256((size_t)N_NODES * H_DIM * 4);
  float* acc1 = (float*)ws;  ws += align256((size_t)N_NODES * H_DIM * 4);
  float* hs2  = (float*)ws;  ws += align256((size_t)N_NODES * D_OUT * 4);
  float* acc2 = (float*)ws;  ws += align256((size_t)N_NODES * D_OUT * 4);
  (void)ws_size; (void)in_sizes; (void)n_in; (void)out_size;

  const int BT = 256;

  // --- normalization ---
  k_init_deg   <<<(N_NODES + BT - 1) / BT, BT, 0, stream>>>(dinv);
  k_accum_deg  <<<(N_EDGES + BT - 1) / BT, BT, 0, stream>>>(dst, dinv);
  k_deg_to_dinv<<<(N_NODES + BT - 1) / BT, BT, 0, stream>>>(dinv);

  // --- layer 1: hs1 = dinv * (x @ W1); acc1 = hs1 (self loop) ---
  // tiles: (50000/16) * (128/16) = 3125*8 = 25000 waves, 4 waves/block
  k_gemm_scale<D_IN, H_DIM><<<25000 / 4, 128, 0, stream>>>(x, W1, dinv, hs1, acc1);
  // scatter: E * 128/4 threads
  k_scatter_edges<H_DIM><<<(N_EDGES * (H_DIM / 4) + BT - 1) / BT, BT, 0, stream>>>(
      src, dst, hs1, acc1);
  // h = relu(dinv*acc1 + b1), reuse hs1 buffer as h
  k_bias_relu<H_DIM><<<(N_NODES * H_DIM + BT - 1) / BT, BT, 0, stream>>>(
      acc1, dinv, b1, hs1);

  // --- layer 2: hs2 = dinv * (h @ W2); acc2 = hs2 ---
  // tiles: 3125 * (64/16) = 12500 waves, 4 waves/block
  k_gemm_scale<H_DIM, D_OUT><<<12500 / 4, 128, 0, stream>>>(hs1, W2, dinv, hs2, acc2);
  k_scatter_edges<D_OUT><<<(N_EDGES * (D_OUT / 4) + BT - 1) / BT, BT, 0, stream>>>(
      src, dst, hs2, acc2);
  k_bias_relu<D_OUT><<<(N_NODES * D_OUT + BT - 1) / BT, BT, 0, stream>>>(
      acc2, dinv, b2, out);
}